// Net_agnn_59768764892011
// MI455X (gfx1250) — compile-verified
//
#include <hip/hip_runtime.h>
#include <math.h>

typedef __attribute__((ext_vector_type(2))) float v2f;
typedef __attribute__((ext_vector_type(8))) float v8f;

// ---------------- float <-> order-preserving uint encoding (for atomic max) --
__device__ __forceinline__ unsigned enc_f32(float f) {
    unsigned u = __float_as_uint(f);
    return (u & 0x80000000u) ? ~u : (u | 0x80000000u);
}
__device__ __forceinline__ float dec_f32(unsigned e) {
    unsigned u = (e & 0x80000000u) ? (e ^ 0x80000000u) : ~e;
    return __uint_as_float(u);
}
#define ENC_NEG_INF 0x007FFFFFu   // enc_f32(-inf)

// ---------------- repack W1 [512,16] f32 -> f32 WMMA B-layout ----------------
// 32-bit B 4x16 fragment per K-chunk of 4: lane l (<16) = column l, K {4t,4t+1};
// lane l (>=16) = column l-16, K {4t+2,4t+3}.  Bw[(t*32+lane)*2 + j].
__global__ void agnn_repack_w1(const float* __restrict__ W1, float* __restrict__ Bw) {
    int i = blockIdx.x * 256 + threadIdx.x;     // 0..4095  (= t*32 + lane)
    if (i >= 128 * 32) return;
    int t = i >> 5, lane = i & 31;
    int n = lane & 15;
    int kb = t * 4 + ((lane < 16) ? 0 : 2);
    Bw[i * 2 + 0] = W1[(kb + 0) * 16 + n];
    Bw[i * 2 + 1] = W1[(kb + 1) * 16 + n];
}

// ---------------- GEMM1: h = relu(x @ W1 + b1), native f32 WMMA --------------
__global__ __launch_bounds__(256)
void agnn_gemm1_relu(const float* __restrict__ x, const float* __restrict__ Bw,
                     const float* __restrict__ b1, float* __restrict__ h, int N) {
    int lane = threadIdx.x & 31;
    int wave = threadIdx.x >> 5;
    int tile = blockIdx.x * 8 + wave;
    int row0 = tile * 16;
    if (row0 >= N) return;

    int rA = row0 + (lane & 15);
    if (rA >= N) rA = N - 1;                    // clamp for tail tiles
    const float* xr = x + (size_t)rA * 512;
    const int koff = (lane < 16) ? 0 : 2;       // 32-bit A layout half-wave K split
    const v2f* Bp  = (const v2f*)Bw;

    v8f c = {};
#pragma unroll 8
    for (int t = 0; t < 128; ++t) {             // K = 512 in chunks of 4
        v2f a = *(const v2f*)(xr + t * 4 + koff);
        v2f b = Bp[t * 32 + lane];
        c = __builtin_amdgcn_wmma_f32_16x16x4_f32(false, a, false, b, (short)0, c, false, false);
    }

    // C/D layout: VGPR r, lanes 0-15 -> M=r, lanes 16-31 -> M=r+8; col = lane&15
    int col   = lane & 15;
    float bia = b1[col];
    int rbase = row0 + ((lane < 16) ? 0 : 8);
    if (row0 + 16 <= N) {                       // full tile: unconditional stores
#pragma unroll
        for (int r = 0; r < 8; ++r) {
            float v = c[r] + bia;
            h[(size_t)(rbase + r) * 16 + col] = v > 0.f ? v : 0.f;
        }
    } else {
#pragma unroll
        for (int r = 0; r < 8; ++r) {
            int row = rbase + r;
            if (row < N) {
                float v = c[r] + bia;
                h[(size_t)row * 16 + col] = v > 0.f ? v : 0.f;
            }
        }
    }
}

// ---------------- per-node inverse L2 norm (eps = 1e-12, like F.normalize) ---
__global__ void agnn_rownorm(const float* __restrict__ f, float* __restrict__ inv, int N) {
    int n = blockIdx.x * 256 + threadIdx.x;
    if (n >= N) return;
    const float4* p = (const float4*)(f + (size_t)n * 16);
    float ss = 0.f;
#pragma unroll
    for (int i = 0; i < 4; ++i) {
        float4 v = p[i];
        ss += v.x * v.x + v.y * v.y + v.z * v.z + v.w * v.w;
    }
    inv[n] = 1.0f / fmaxf(sqrtf(ss), 1e-12f);
}

// ---------------- init segment max / sum / output accumulator ----------------
__global__ void agnn_conv_init(unsigned* __restrict__ m, float* __restrict__ s,
                               float* __restrict__ out, int N) {
    int n = blockIdx.x * 256 + threadIdx.x;
    if (n >= N) return;
    m[n] = ENC_NEG_INF;
    s[n] = 0.f;
    float4 z = make_float4(0.f, 0.f, 0.f, 0.f);
    float4* o = (float4*)(out + (size_t)n * 16);
    o[0] = z; o[1] = z; o[2] = z; o[3] = z;
}

// ---------------- pass 1: per-edge cosine logit + segment max ----------------
__global__ void agnn_edge_logits(const int* __restrict__ srcI, const int* __restrict__ dstI,
                                 const float* __restrict__ f, const float* __restrict__ inv,
                                 const float* __restrict__ betap, float* __restrict__ lg,
                                 unsigned* __restrict__ m, int E, int T) {
    int i = blockIdx.x * 256 + threadIdx.x;
    if (i >= T) return;
    int s = (i < E) ? srcI[i] : (i - E);   // appended self-loops
    int d = (i < E) ? dstI[i] : (i - E);
    const float4* ps = (const float4*)(f + (size_t)s * 16);
    const float4* pd = (const float4*)(f + (size_t)d * 16);
    float dot = 0.f;
#pragma unroll
    for (int k = 0; k < 4; ++k) {
        float4 a = ps[k], b = pd[k];
        dot += a.x * b.x + a.y * b.y + a.z * b.z + a.w * b.w;
    }
    float beta  = betap ? betap[0] : 1.0f;
    float logit = beta * dot * inv[s] * inv[d];
    lg[i] = logit;
    atomicMax(&m[d], enc_f32(logit));
}

// ---------------- pass 2: e = exp(l - m[dst]); segment sum -------------------
__global__ void agnn_edge_exp(const int* __restrict__ dstI, float* __restrict__ lg,
                              const unsigned* __restrict__ m, float* __restrict__ s,
                              int E, int T) {
    int i = blockIdx.x * 256 + threadIdx.x;
    if (i >= T) return;
    int d = (i < E) ? dstI[i] : (i - E);
    float md = dec_f32(m[d]);
    if (!isfinite(md)) md = 0.f;           // jnp.where(isfinite(m), m, 0)
    float e = __expf(lg[i] - md);
    lg[i] = e;
    atomicAdd(&s[d], e);
}

// ---------------- pass 3: out[dst] += (e / (s+eps)) * x[src] -----------------
__global__ void agnn_edge_scatter(const int* __restrict__ srcI, const int* __restrict__ dstI,
                                  const float* __restrict__ f, const float* __restrict__ lg,
                                  const float* __restrict__ s, float* __restrict__ out,
                                  int E, int T) {
    int i = blockIdx.x * 256 + threadIdx.x;
    if (i >= T) return;
    int sn = (i < E) ? srcI[i] : (i - E);
    int dn = (i < E) ? dstI[i] : (i - E);
    float alpha = lg[i] / (s[dn] + 1e-16f);
    const float* xs = f + (size_t)sn * 16;
    float* o = out + (size_t)dn * 16;
#pragma unroll
    for (int c = 0; c < 16; ++c)
        atomicAdd(&o[c], alpha * xs[c]);
}

// ---------------- head: logits = h2 @ W2 + b2; log_softmax -------------------
__global__ void agnn_head(const float* __restrict__ h2, const float* __restrict__ W2,
                          const float* __restrict__ b2, float* __restrict__ out, int N) {
    int n = blockIdx.x * 256 + threadIdx.x;
    if (n >= N) return;
    float z[16];
#pragma unroll
    for (int c = 0; c < 16; ++c) z[c] = b2[c];
#pragma unroll
    for (int k = 0; k < 16; ++k) {
        float hk = h2[(size_t)n * 16 + k];
#pragma unroll
        for (int c = 0; c < 16; ++c) z[c] += hk * W2[k * 16 + c];
    }
    float mx = z[0];
#pragma unroll
    for (int c = 1; c < 16; ++c) mx = fmaxf(mx, z[c]);
    float se = 0.f;
#pragma unroll
    for (int c = 0; c < 16; ++c) se += __expf(z[c] - mx);
    float lse = mx + logf(se);
#pragma unroll
    for (int c = 0; c < 16; ++c) out[(size_t)n * 16 + c] = z[c] - lse;
}

// ---------------------------------------------------------------------------
extern "C" void kernel_launch(void* const* d_in, const int* in_sizes, int n_in,
                              void* d_out, int out_size, void* d_ws, size_t ws_size,
                              hipStream_t stream) {
    const float* x     = (const float*)d_in[0];
    const int*   eidx  = (const int*)d_in[1];       // [2,E]: src row then dst row
    const float* W1    = (const float*)d_in[2];
    const float* b1    = (const float*)d_in[3];
    const float* W2    = (const float*)d_in[4];
    const float* b2    = (const float*)d_in[5];
    const float* beta2 = (const float*)d_in[6];

    const int N = in_sizes[0] / 512;
    const int E = in_sizes[1] / 2;
    const int T = E + N;                            // edges + self-loops

    float* out_ls = (float*)d_out;                  // log_softmax  [N,16]
    float* x1     = out_ls + (size_t)N * 16;        // conv1 output [N,16] (2nd tuple elem)

    // carve workspace (256B aligned slices)
    char* w = (char*)d_ws;
    auto carve = [&](size_t bytes) -> void* {
        void* p = (void*)w;
        w += (bytes + 255) & ~(size_t)255;
        return p;
    };
    float*    Bw  = (float*)carve(128 * 32 * 2 * sizeof(float));   // repacked W1 (32 KB)
    float*    h   = (float*)carve((size_t)N * 16 * sizeof(float));
    float*    h2  = (float*)carve((size_t)N * 16 * sizeof(float));
    float*    inv = (float*)carve((size_t)N * sizeof(float));
    unsigned* m   = (unsigned*)carve((size_t)N * sizeof(unsigned));
    float*    s   = (float*)carve((size_t)N * sizeof(float));
    float*    lg  = (float*)carve((size_t)T * sizeof(float));

    const int tiles = (N + 15) / 16;
    dim3 bn((N + 255) / 256), be((T + 255) / 256);

    // h = relu(x @ W1 + b1)    [single pass over the 205 MB x matrix, f32 WMMA]
    agnn_repack_w1<<<16, 256, 0, stream>>>(W1, Bw);
    agnn_gemm1_relu<<<(tiles + 7) / 8, 256, 0, stream>>>(x, Bw, b1, h, N);

    // conv1: beta = 1.0, features = h, output = x1 (directly into d_out)
    agnn_rownorm<<<bn, 256, 0, stream>>>(h, inv, N);
    agnn_conv_init<<<bn, 256, 0, stream>>>(m, s, x1, N);
    agnn_edge_logits<<<be, 256, 0, stream>>>(eidx, eidx + E, h, inv, nullptr, lg, m, E, T);
    agnn_edge_exp<<<be, 256, 0, stream>>>(eidx + E, lg, m, s, E, T);
    agnn_edge_scatter<<<be, 256, 0, stream>>>(eidx, eidx + E, h, lg, s, x1, E, T);

    // conv2: beta = beta2[0], features = x1, output = h2
    agnn_rownorm<<<bn, 256, 0, stream>>>(x1, inv, N);
    agnn_conv_init<<<bn, 256, 0, stream>>>(m, s, h2, N);
    agnn_edge_logits<<<be, 256, 0, stream>>>(eidx, eidx + E, x1, inv, beta2, lg, m, E, T);
    agnn_edge_exp<<<be, 256, 0, stream>>>(eidx + E, lg, m, s, E, T);
    agnn_edge_scatter<<<be, 256, 0, stream>>>(eidx, eidx + E, x1, lg, s, h2, E, T);

    // log_softmax(h2 @ W2 + b2)
    agnn_head<<<bn, 256, 0, stream>>>(h2, W2, b2, out_ls, N);
}